// Divergence2d_11373073399939
// MI455X (gfx1250) — compile-verified
//
#include <hip/hip_runtime.h>

// Divergence2d: (16,16,512,512) f32 -> (16,2,514,514) f32
// HBM-bound stencil (~290 MB traffic, ~12.5 us floor @ 23.3 TB/s).
// Staged through LDS with CDNA5 async global->LDS copies (ASYNCcnt path).

#define IN_H   512
#define IN_W   512
#define OUT_H  514
#define OUT_W  514
#define NCH    16
#define NROWS  24                 // 8 x-rows (oy-1), 8 y-top (oy-2), 8 y-bot (oy)
#define VECS_PER_ROW (IN_W / 4)   // 128 float4 per row

// Async copy of 16 bytes global -> LDS (per lane). Tracked by ASYNCcnt.
__device__ __forceinline__ void async_copy_b128(unsigned lds_byte_addr,
                                                const float* gaddr) {
  asm volatile("global_load_async_to_lds_b128 %0, %1, off"
               :
               : "v"(lds_byte_addr), "v"(gaddr)
               : "memory");
}

__device__ __forceinline__ void wait_async_zero() {
  asm volatile("s_wait_asynccnt 0x0" ::: "memory");
}

// slot -> (channel, y) mapping:
//   slot 0..7  : channels {0..3, 8..11}  at row oy-1  (x-derivative rows)
//   slot 8..15 : channels {4..7, 12..15} at row oy-2  (y-derivative top)
//   slot 16..23: channels {4..7, 12..15} at row oy    (y-derivative bottom)
__device__ __forceinline__ void slot_decode(int slot, int oy, int& c, int& y) {
  if (slot < 8)       { c = (slot < 4) ? slot : slot + 4;                     y = oy - 1; }
  else if (slot < 16) { const int k = slot - 8;  c = (k < 4) ? k + 4 : k + 8; y = oy - 2; }
  else                { const int k = slot - 16; c = (k < 4) ? k + 4 : k + 8; y = oy;     }
}

__global__ __launch_bounds__(256)
void divergence2d_kernel(const float* __restrict__ x,
                         const float* __restrict__ lam1x,
                         float* __restrict__ out) {
  __shared__ float smem[NROWS * IN_W];   // 48 KB
  const int tid = threadIdx.x;
  const int oy  = blockIdx.x % OUT_H;    // oy-major: adjacent blocks share rows (L2 reuse)
  const int n   = blockIdx.x / OUT_H;

  // ---- stage 24 input rows into LDS: exactly 12 uniform passes of 2 slots ----
  const int vec  = tid & (VECS_PER_ROW - 1);
  const int half = tid >> 7;             // wave-uniform (0 for waves 0..3, 1 for 4..7)
#pragma unroll
  for (int s = 0; s < 12; ++s) {
    const int slot = 2 * s + half;
    int c, y;
    slot_decode(slot, oy, c, y);
    float* dstp = &smem[slot * IN_W + vec * 4];
    if ((unsigned)y < (unsigned)IN_H) {
      const float* src =
          x + (((size_t)(n * NCH + c) * IN_H + (size_t)y) * IN_W + (size_t)(vec * 4));
      // low 32 bits of a generic pointer into __shared__ == LDS byte address
      async_copy_b128((unsigned)(uintptr_t)dstp, src);
    } else {
      *(float4*)dstp = make_float4(0.f, 0.f, 0.f, 0.f);   // zero padding row (ds_store_b128)
    }
  }
  wait_async_zero();   // this wave's async LDS writes landed
  __syncthreads();     // all waves' writes (async + ds zero-fill) visible

  // per-channel lambda for the first x-derivative group (0.25 in setup)
  float lam[4];
#pragma unroll
  for (int k = 0; k < 4; ++k) lam[k] = lam1x[k];

  const float* sX  = smem;               // 8 rows: x-deriv source row (oy-1)
  const float* sYt = smem + 8  * IN_W;   // 8 rows: y-deriv top (oy-2)
  const float* sYb = smem + 16 * IN_W;   // 8 rows: y-deriv bottom (oy)

  float* out1 = out + (((size_t)n * 2 + 0) * OUT_H + (size_t)oy) * OUT_W;
  float* out2 = out + (((size_t)n * 2 + 1) * OUT_H + (size_t)oy) * OUT_W;

  // ---- interior columns ox in [2, 511]: all taps in range, zero predication ----
  for (int ox = 2 + tid; ox < IN_W; ox += 256) {
    float o1 = 0.f, o2 = 0.f;
#pragma unroll
    for (int k = 0; k < 4; ++k) {
      o1 += lam[k] * (sX[k * IN_W + ox] - sX[k * IN_W + ox - 2]);
      o2 += sX[(k + 4) * IN_W + ox] - sX[(k + 4) * IN_W + ox - 2];
      o1 += sYt[k * IN_W + ox - 1]       - sYb[k * IN_W + ox - 1];
      o2 += sYt[(k + 4) * IN_W + ox - 1] - sYb[(k + 4) * IN_W + ox - 1];
    }
    __builtin_nontemporal_store(o1, &out1[ox]);   // streamed output: TH=NT
    __builtin_nontemporal_store(o2, &out2[ox]);
  }

  // ---- edge columns ox in {0, 1, 512, 513}: predicated taps ----
  if (tid < 4) {
    const int ox = (tid < 2) ? tid : (IN_W + tid - 2);   // 0,1,512,513
    const bool vR = (ox <= IN_W - 1);
    const bool vL = (ox >= 2);
    const bool vM = (ox >= 1) && (ox <= IN_W);
    float o1 = 0.f, o2 = 0.f;
#pragma unroll
    for (int k = 0; k < 4; ++k) {
      const float r1 = vR ? sX[k * IN_W + ox] : 0.f;
      const float l1 = vL ? sX[k * IN_W + ox - 2] : 0.f;
      const float r2 = vR ? sX[(k + 4) * IN_W + ox] : 0.f;
      const float l2 = vL ? sX[(k + 4) * IN_W + ox - 2] : 0.f;
      o1 += lam[k] * (r1 - l1);
      o2 += r2 - l2;
      if (vM) {
        o1 += sYt[k * IN_W + ox - 1]       - sYb[k * IN_W + ox - 1];
        o2 += sYt[(k + 4) * IN_W + ox - 1] - sYb[(k + 4) * IN_W + ox - 1];
      }
    }
    __builtin_nontemporal_store(o1, &out1[ox]);
    __builtin_nontemporal_store(o2, &out2[ox]);
  }
}

extern "C" void kernel_launch(void* const* d_in, const int* in_sizes, int n_in,
                              void* d_out, int out_size, void* d_ws, size_t ws_size,
                              hipStream_t stream) {
  (void)in_sizes; (void)n_in; (void)out_size; (void)d_ws; (void)ws_size;
  const float* x     = (const float*)d_in[0];
  const float* lam1x = (const float*)d_in[1];   // (1,4,1,1); lam2x/1y/2y unused by reference
  float* out = (float*)d_out;

  const int nblocks = 16 * OUT_H;               // one block per (n, oy) output row pair
  divergence2d_kernel<<<nblocks, 256, 0, stream>>>(x, lam1x, out);
}